// ThetaLoss_52518860096008
// MI455X (gfx1250) — compile-verified
//
#include <hip/hip_runtime.h>
#include <cstdint>

typedef __attribute__((ext_vector_type(2))) float v2f;
typedef __attribute__((ext_vector_type(8))) float v8f;

#define K1_BLOCKS  1024
#define K1_THREADS 256
#define K1_WAVES   8

#if defined(__has_builtin)
#if __has_builtin(__builtin_amdgcn_wmma_f32_16x16x4_f32)
#define HAVE_WMMA_16X16X4_F32 1
#endif
#endif

// ---------- gfx1250 async global->LDS (ASYNCcnt path) ----------
__device__ __forceinline__ void async_ld_b128(unsigned lds_off, const void* gaddr) {
  asm volatile("global_load_async_to_lds_b128 %0, %1, off"
               :: "v"(lds_off), "v"((unsigned long long)gaddr)
               : "memory");
}
__device__ __forceinline__ void wait_async(bool keep6) {
  if (keep6) asm volatile("s_wait_asynccnt 0x6" ::: "memory");
  else       asm volatile("s_wait_asynccnt 0x0" ::: "memory");
}

// ---------- libcall-free acos (minimax, |err| ~1e-7 rad) ----------
__device__ __forceinline__ float acos_fast(float x) {
  float ax = fabsf(x);
  float p = -0.0012624911f;
  p = p * ax + 0.0066700901f;
  p = p * ax - 0.0170881256f;
  p = p * ax + 0.0308918810f;
  p = p * ax - 0.0501743046f;
  p = p * ax + 0.0889789874f;
  p = p * ax - 0.2145988016f;
  p = p * ax + 1.5707963050f;
  float r = __builtin_sqrtf(1.0f - ax) * p;
  return (x >= 0.0f) ? r : (3.14159265358979f - r);
}

// ---------- deterministic wave32 reduction of two scalars ----------
__device__ __forceinline__ void wave_reduce2(float& x, float& y) {
#ifdef HAVE_WMMA_16X16X4_F32
  // Stage 1: A[m,0]=x_m, A[m,1]=y_m, A[m,2]=x_{m+16}, A[m,3]=y_{m+16}.
  // Selector B picks x (K=0,2) or y (K=1,3): D[m,n] = x_m + x_{m+16} (resp. y).
  v2f a;  a.x = x;    a.y = y;
  v2f bx; bx.x = 1.f; bx.y = 0.f;   // vgpr0 -> K=0 (lanes<16) / K=2 (lanes>=16)
  v2f by; by.x = 0.f; by.y = 1.f;   // vgpr1 -> K=1 / K=3
  v8f z = {};
  v8f dx = __builtin_amdgcn_wmma_f32_16x16x4_f32(false, a, false, bx, (short)0, z, false, false);
  v8f dy = __builtin_amdgcn_wmma_f32_16x16x4_f32(false, a, false, by, (short)0, z, false, false);
  // Stage 2: A = ones(16x4); B pairs (dx[p],dx[p+1]) cover rows {p,p+1,p+8,p+9};
  // accumulating p=0,2,4,6 sums all 16 rows -> every element of C = full wave sum.
  v2f ones; ones.x = 1.f; ones.y = 1.f;
  v8f cx = {}, cy = {};
#pragma unroll
  for (int p = 0; p < 8; p += 2) {
    v2f bpx; bpx.x = dx[p]; bpx.y = dx[p + 1];
    v2f bpy; bpy.x = dy[p]; bpy.y = dy[p + 1];
    cx = __builtin_amdgcn_wmma_f32_16x16x4_f32(false, ones, false, bpx, (short)0, cx, false, false);
    cy = __builtin_amdgcn_wmma_f32_16x16x4_f32(false, ones, false, bpy, (short)0, cy, false, false);
  }
  x = cx[0];
  y = cy[0];
#else
#pragma unroll
  for (int off = 16; off > 0; off >>= 1) {
    x += __shfl_xor(x, off, 32);
    y += __shfl_xor(y, off, 32);
  }
#endif
}

// ---------- per-item math: MSE + closed-form singular values of 3x3 ----------
__device__ __forceinline__ void accum_item(float4 t0, float4 t1, float4 t2,
                                           float4 a0, float4 a1, float4 a2,
                                           float& ortho, float& mse) {
  float d, m = 0.f;
  d = t0.x - a0.x; m += d * d;  d = t0.y - a0.y; m += d * d;
  d = t0.z - a0.z; m += d * d;  d = t0.w - a0.w; m += d * d;
  d = t1.x - a1.x; m += d * d;  d = t1.y - a1.y; m += d * d;
  d = t1.z - a1.z; m += d * d;  d = t1.w - a1.w; m += d * d;
  d = t2.x - a2.x; m += d * d;  d = t2.y - a2.y; m += d * d;
  d = t2.z - a2.z; m += d * d;  d = t2.w - a2.w; m += d * d;
  mse += m;

  // A = W^T W (symmetric PSD), W rows = (tR.x, tR.y, tR.z)
  float a00 = t0.x * t0.x + t1.x * t1.x + t2.x * t2.x;
  float a11 = t0.y * t0.y + t1.y * t1.y + t2.y * t2.y;
  float a22 = t0.z * t0.z + t1.z * t1.z + t2.z * t2.z;
  float a01 = t0.x * t0.y + t1.x * t1.y + t2.x * t2.y;
  float a02 = t0.x * t0.z + t1.x * t1.z + t2.x * t2.z;
  float a12 = t0.y * t0.z + t1.y * t1.z + t2.y * t2.z;

  // Trigonometric closed-form eigenvalues of a symmetric 3x3
  float q  = (a00 + a11 + a22) * (1.0f / 3.0f);
  float p1 = a01 * a01 + a02 * a02 + a12 * a12;
  float b00 = a00 - q, b11 = a11 - q, b22 = a22 - q;
  float p2 = b00 * b00 + b11 * b11 + b22 * b22 + 2.0f * p1;
  float p  = __builtin_sqrtf(p2 * (1.0f / 6.0f));
  float ip = (p > 0.0f) ? (1.0f / p) : 0.0f;
  b00 *= ip; b11 *= ip; b22 *= ip;
  float c01 = a01 * ip, c02 = a02 * ip, c12 = a12 * ip;
  float detB = b00 * (b11 * b22 - c12 * c12)
             - c01 * (c01 * b22 - c12 * c02)
             + c02 * (c01 * c12 - b11 * c02);
  float r = fminf(fmaxf(0.5f * detB, -1.0f), 1.0f);
  float phi = acos_fast(r) * (1.0f / 3.0f);
  float twop = 2.0f * p;
  float l1 = q + twop * __builtin_cosf(phi);
  float l3 = q + twop * __builtin_cosf(phi + 2.0943951023931953f);  // +2pi/3
  float l2 = 3.0f * q - l1 - l3;

  float s, s2, o = -6.0f;
  s = __builtin_sqrtf(fmaxf(l1, 0.0f)) + 1e-6f; s2 = s * s; o += s2 + 1.0f / s2;
  s = __builtin_sqrtf(fmaxf(l2, 0.0f)) + 1e-6f; s2 = s * s; o += s2 + 1.0f / s2;
  s = __builtin_sqrtf(fmaxf(l3, 0.0f)) + 1e-6f; s2 = s * s; o += s2 + 1.0f / s2;
  ortho += o;
}

// ---------- kernel 1: streaming partials (async->LDS double buffered per wave) ----------
__global__ __launch_bounds__(K1_THREADS)
void theta_partials(const float* __restrict__ theta, const float* __restrict__ affine,
                    float* __restrict__ ws, int nItems, int tilesPerBlock) {
  // per wave, per buffer: 32 items * (12 theta + 12 affine) floats = 768 floats = 3 KB
  __shared__ float lds[2][K1_WAVES][768];
  __shared__ float redO[K1_WAVES], redM[K1_WAVES];

  const int tid  = threadIdx.x;
  const int w    = tid >> 5;
  const int lane = tid & 31;
  const long blockBase = (long)blockIdx.x * (long)tilesPerBlock * K1_THREADS;

  const float4* t4 = (const float4*)theta;
  const float4* a4 = (const float4*)affine;

  float ortho = 0.0f, mse = 0.0f;

  auto fullTile = [&](int j) -> bool {
    long base = blockBase + (long)j * K1_THREADS + (long)(w * 32);
    return base + 32 <= (long)nItems;
  };
  auto issue = [&](int j, int buf) {  // 6 async b128 loads (wave-contiguous 3 KB)
    long base   = blockBase + (long)j * K1_THREADS + (long)(w * 32);
    long f4base = base * 3;  // 3 float4 per item
    unsigned ldsT = (unsigned)(uintptr_t)&lds[buf][w][0];
    unsigned ldsA = ldsT + 384u * 4u;
#pragma unroll
    for (int rr = 0; rr < 3; ++rr) {
      int k = rr * 32 + lane;
      async_ld_b128(ldsT + (unsigned)(k * 16), (const void*)(t4 + f4base + k));
      async_ld_b128(ldsA + (unsigned)(k * 16), (const void*)(a4 + f4base + k));
    }
  };

  if (fullTile(0)) issue(0, 0);
  for (int j = 0; j < tilesPerBlock; ++j) {
    const bool fj = fullTile(j);
    const bool fn = (j + 1 < tilesPerBlock) && fullTile(j + 1);
    if (fn) issue(j + 1, (j + 1) & 1);
    if (fj) {
      wait_async(fn);  // <=6 outstanding -> previous tile's copies complete (in-order)
      const int buf = j & 1;
      const float* L  = &lds[buf][w][lane * 12];
      const float* LA = &lds[buf][w][384 + lane * 12];
      accum_item(*(const float4*)(L + 0), *(const float4*)(L + 4), *(const float4*)(L + 8),
                 *(const float4*)(LA + 0), *(const float4*)(LA + 4), *(const float4*)(LA + 8),
                 ortho, mse);
    } else {  // tail: guarded direct loads (async counter already drained)
      long idx = blockBase + (long)j * K1_THREADS + tid;
      if (idx < (long)nItems) {
        long f = idx * 3;
        accum_item(t4[f], t4[f + 1], t4[f + 2], a4[f], a4[f + 1], a4[f + 2], ortho, mse);
      }
    }
  }

  wave_reduce2(ortho, mse);            // uniform control flow, EXEC all-ones
  if (lane == 0) { redO[w] = ortho; redM[w] = mse; }
  __syncthreads();
  if (tid == 0) {
    float o = 0.f, m = 0.f;
#pragma unroll
    for (int i = 0; i < K1_WAVES; ++i) { o += redO[i]; m += redM[i]; }
    ws[2 * blockIdx.x]     = o;
    ws[2 * blockIdx.x + 1] = m;
  }
}

// ---------- kernel 2: deterministic finalize (1 wave) ----------
__global__ __launch_bounds__(32)
void theta_finalize(const float* __restrict__ ws, float* __restrict__ out,
                    int nblocks, int nItems) {
  const int lane = threadIdx.x;
  float o = 0.f, m = 0.f;
  for (int i = lane; i < nblocks; i += 32) { o += ws[2 * i]; m += ws[2 * i + 1]; }
  wave_reduce2(o, m);
  if (lane == 0) {
    float invN = 1.0f / (float)nItems;
    out[0] = o * invN;                    // mean over batch
    out[1] = m * (invN * (1.0f / 12.0f)); // mean over batch*3*4 elements
  }
}

extern "C" void kernel_launch(void* const* d_in, const int* in_sizes, int n_in,
                              void* d_out, int out_size, void* d_ws, size_t ws_size,
                              hipStream_t stream) {
  const float* theta  = (const float*)d_in[0];
  const float* affine = (const float*)d_in[1];
  float* out = (float*)d_out;
  float* ws  = (float*)d_ws;

  const int nItems = in_sizes[0] / 12;
  int blocks = K1_BLOCKS;
  if ((size_t)blocks * 2 * sizeof(float) > ws_size)
    blocks = (int)(ws_size / (2 * sizeof(float)));
  if (blocks < 1) blocks = 1;
  long perPass = (long)blocks * K1_THREADS;
  int tiles = (int)(((long)nItems + perPass - 1) / perPass);
  if (tiles < 1) tiles = 1;

  theta_partials<<<blocks, K1_THREADS, 0, stream>>>(theta, affine, ws, nItems, tiles);
  theta_finalize<<<1, 32, 0, stream>>>(ws, out, blocks, nItems);
}